// CostVolume_4939212390829
// MI455X (gfx1250) — compile-verified
//
#include <hip/hip_runtime.h>
#include <stdint.h>

// ---- Problem constants (from reference) ----
#define BB 2
#define CC 32
#define HH 136
#define WW 240
#define DD 48
#define CLAMP_V 1000.0f

typedef float v4f  __attribute__((ext_vector_type(4)));
typedef unsigned int u32x4 __attribute__((ext_vector_type(4)));
typedef int i32x4  __attribute__((ext_vector_type(4)));
typedef int i32x8  __attribute__((ext_vector_type(8)));

#if defined(__gfx1250__) && __has_builtin(__builtin_amdgcn_tensor_load_to_lds)
#define USE_TDM 1
#else
#define USE_TDM 0
#endif

#if USE_TDM
// Issue one TDM load of a contiguous row of WW floats (960 B) into LDS.
// D# layout per CDNA5 ISA ch.8 (CDNA5 ISA):
//  group0: [1:0]=count=1, [63:32]=lds_addr, [120:64]=global_addr, [127:126]=type=2
//  group1: [17:16]=data_size (2 => 4B), [79:48]=tensor_dim0, [111:80]=tensor_dim1,
//          [127:112]=tile_dim0, [143:128]=tile_dim1, [207:160]=tensor_dim0_stride,
//          [255:208]=tensor_dim1_stride
//  groups 2/3 (and trailing group on the 6-arg clang-23 form): zero (<=2D tensor)
__device__ __forceinline__ void tdm_load_row(const float* gsrc, unsigned lds_byte_off) {
    const uint64_t ga = (uint64_t)(uintptr_t)gsrc;

    u32x4 g0;
    g0[0] = 1u;                                               // count=1, user mode
    g0[1] = lds_byte_off;                                     // lds_addr
    g0[2] = (unsigned)ga;                                     // global_addr[31:0]
    g0[3] = ((unsigned)(ga >> 32) & 0x01FFFFFFu) | (2u << 30);// global_addr[56:32] | type=2

    i32x8 g1;
    g1[0] = (2 << 16);                 // data_size = 2 (4 bytes); wg_mask=0
    g1[1] = (int)((WW & 0xFFFF) << 16);// tensor_dim0[15:0] at bits 63:48
    g1[2] = (int)((WW >> 16) | (1 << 16)); // tensor_dim0[31:16] | tensor_dim1=1
    g1[3] = (int)(WW << 16);           // tile_dim0 = WW at bits 127:112
    g1[4] = 1;                         // tile_dim1 = 1, tile_dim2 = 0
    g1[5] = WW;                        // tensor_dim0_stride[31:0]
    g1[6] = (int)(WW << 16);           // tensor_dim1_stride[15:0] at bits 223:208
    g1[7] = 0;

    i32x4 z4 = {0, 0, 0, 0};
    i32x8 z8 = {0, 0, 0, 0, 0, 0, 0, 0};
    // clang-23 / therock-10.0 form: (g0, g1, g2, g3, extra, cpol)
    __builtin_amdgcn_tensor_load_to_lds(g0, g1, z4, z4, z8, 0);
}
#endif

__global__ __launch_bounds__(256) void
cost_volume_kernel(const float* __restrict__ left,
                   const float* __restrict__ right,
                   float* __restrict__ out) {
    __shared__ __align__(16) float sl[WW];
    __shared__ __align__(16) float sr[WW];

    const int row = blockIdx.x;          // flat over B*C*H
    const int h   = row % HH;
    const int bc  = row / HH;            // b*C + c
    const float* lrow = left  + (size_t)row * WW;
    const float* rrow = right + (size_t)row * WW;

#if USE_TDM
    if (threadIdx.x == 0) {
        // generic LDS pointer: low 32 bits are the LDS byte offset
        tdm_load_row(lrow, (unsigned)(uintptr_t)&sl[0]);
        tdm_load_row(rrow, (unsigned)(uintptr_t)&sr[0]);
    }
#if __has_builtin(__builtin_amdgcn_s_wait_tensorcnt)
    __builtin_amdgcn_s_wait_tensorcnt(0);
#else
    asm volatile("s_wait_tensorcnt 0x0" ::: "memory");
#endif
#else
    for (int i = threadIdx.x; i < WW; i += blockDim.x) {
        sl[i] = lrow[i];
        sr[i] = rrow[i];
    }
#endif
    __syncthreads();

    // d-major over quads of w: lanes with the same d write consecutive 16B
    // chunks -> coalesced 128-bit non-temporal stores.
    constexpr int NQ = WW / 4;           // 60 quads per output row
    const int tid = threadIdx.x;

    for (int idx = tid; idx < DD * NQ; idx += 256) {
        const int d  = idx / NQ;
        const int q  = idx - d * NQ;
        const int w0 = q * 4;

        const v4f lv = *(const v4f*)(&sl[w0]);
        v4f res;
#pragma unroll
        for (int j = 0; j < 4; ++j) {
            const int w = w0 + j;
            const int s = w - d;
            float v = (s >= 0) ? (lv[j] * sr[s]) : 0.0f;
            v = fminf(fmaxf(v, -CLAMP_V), CLAMP_V);
            res[j] = v;
        }

        const size_t obase = ((size_t)(bc * DD + d) * HH + h) * (size_t)WW + (size_t)w0;
        // Output is 401 MB streamed once and never re-read: non-temporal store.
        __builtin_nontemporal_store(res, (v4f*)(out + obase));
    }
}

extern "C" void kernel_launch(void* const* d_in, const int* in_sizes, int n_in,
                              void* d_out, int out_size, void* d_ws, size_t ws_size,
                              hipStream_t stream) {
    const float* left  = (const float*)d_in[0];
    const float* right = (const float*)d_in[1];
    float* out = (float*)d_out;

    const dim3 grid(BB * CC * HH);   // 8704 row-workgroups
    const dim3 block(256);           // 8 wave32 per WGP-resident block
    cost_volume_kernel<<<grid, block, 0, stream>>>(left, right, out);
}